// GATNet_35381940584639
// MI455X (gfx1250) — compile-verified
//
#include <hip/hip_runtime.h>
#include <math.h>

// ---------------------------------------------------------------------------
// GATNet forward for MI455X (gfx1250, wave32).
//  - Dense h = x@W via V_WMMA_F32_16X16X4_F32, W staged in LDS, templated dims
//  - CSR-by-dst built once; softmax-aggregation fused, atomic-free, float4
//  - Training-mode BatchNorm via two-pass column reduction
// ---------------------------------------------------------------------------

#define NHEADS 4

typedef __attribute__((ext_vector_type(2))) float v2f;
typedef __attribute__((ext_vector_type(8))) float v8f;

static __device__ __forceinline__ float lrelu(float v) {
  return v > 0.f ? v : 0.2f * v;
}

// ---------------------------------------------------------------------------
// GEMM: Y[n, DOUT] = X[n, K] @ W[K, DOUT], fp32, WMMA 16x16x4.
// One wave computes a full 16 x DOUT row panel; W lives in LDS.
// nRows % 16 == 0 (100000 = 6250 * 16).
// ---------------------------------------------------------------------------
template <int K, int DOUT>
__global__ void gemm_wmma_f32(const float* __restrict__ X,
                              const float* __restrict__ W,
                              float* __restrict__ Y, int nRows) {
  constexpr int NT = DOUT / 16;
  __shared__ float sW[K * DOUT];
  for (int idx = threadIdx.x; idx < K * DOUT; idx += blockDim.x)
    sW[idx] = W[idx];
  __syncthreads();

  const int lane = threadIdx.x & 31;
  const int wave = threadIdx.x >> 5;
  const int mt = blockIdx.x * (blockDim.x >> 5) + wave;  // wave-uniform
  if (mt * 16 >= nRows) return;                          // uniform exit

  const int half = lane >> 4;  // 0: lanes 0-15, 1: lanes 16-31
  const int l15  = lane & 15;

  v8f acc[NT];
#pragma unroll
  for (int nt = 0; nt < NT; ++nt) acc[nt] = (v8f){};

  const float* xrow = X + (size_t)(mt * 16 + l15) * K;
  for (int kk = 0; kk < K; kk += 4) {
    // A 16x4 (MxK): lanes 0-15 carry K=kk,kk+1; lanes 16-31 carry K=kk+2,kk+3
    const int ka = kk + half * 2;
    v2f a; a.x = xrow[ka]; a.y = xrow[ka + 1];
#pragma unroll
    for (int nt = 0; nt < NT; ++nt) {
      v2f b;
      b.x = sW[ka * DOUT + nt * 16 + l15];
      b.y = sW[(ka + 1) * DOUT + nt * 16 + l15];
      acc[nt] = __builtin_amdgcn_wmma_f32_16x16x4_f32(
          false, a, false, b, (short)0, acc[nt], false, false);
    }
  }
  // C/D: VGPR r -> M = r (lanes 0-15) or r+8 (lanes 16-31), N = l15
#pragma unroll
  for (int nt = 0; nt < NT; ++nt) {
#pragma unroll
    for (int r = 0; r < 8; ++r) {
      const int row = mt * 16 + r + half * 8;
      Y[(size_t)row * DOUT + nt * 16 + l15] = acc[nt][r];
    }
  }
}

// ---------------------------------------------------------------------------
// Per-node attention logits: als[n,h] = <h[n,h,:], a_src[h,:]> (and dst).
// ---------------------------------------------------------------------------
__global__ void gat_scores(const float* __restrict__ H,
                           const float* __restrict__ asrc,
                           const float* __restrict__ adst,
                           float* __restrict__ als, float* __restrict__ ald,
                           int n, int C) {
  const int i = blockIdx.x * blockDim.x + threadIdx.x;
  if (i >= n * NHEADS) return;
  const int node = i >> 2, hd = i & 3;
  const float4* hp = (const float4*)(H + (size_t)node * NHEADS * C + hd * C);
  const float4* ap = (const float4*)(asrc + hd * C);
  const float4* dp = (const float4*)(adst + hd * C);
  float ss = 0.f, sd = 0.f;
  for (int c = 0; c < C / 4; ++c) {
    const float4 v = hp[c], a = ap[c], d = dp[c];
    ss += v.x * a.x + v.y * a.y + v.z * a.z + v.w * a.w;
    sd += v.x * d.x + v.y * d.y + v.z * d.z + v.w * d.w;
  }
  als[i] = ss;
  ald[i] = sd;
}

// ---------------------------------------------------------------------------
// CSR build: degree count, 3-kernel exclusive scan (LDS), scatter.
// Self loops appended after the E real edges.
// ---------------------------------------------------------------------------
__global__ void deg_count(const int* __restrict__ ei, int E, int total,
                          int* __restrict__ cnt) {
  const int e = blockIdx.x * blockDim.x + threadIdx.x;
  if (e >= total) return;
  const int d = (e < E) ? ei[E + e] : e - E;
  atomicAdd(&cnt[d], 1);
}

// Block partial sums over chunks of 1024 (256 threads x 4).
__global__ void scan_partial(const int* __restrict__ cnt, int n,
                             int* __restrict__ bsum) {
  __shared__ int sd[256];
  const int t = threadIdx.x;
  const int base = blockIdx.x * 1024 + t * 4;
  int s = 0;
#pragma unroll
  for (int j = 0; j < 4; ++j)
    if (base + j < n) s += cnt[base + j];
  sd[t] = s;
  __syncthreads();
  for (int off = 128; off > 0; off >>= 1) {
    if (t < off) sd[t] += sd[t + off];
    __syncthreads();
  }
  if (t == 0) bsum[blockIdx.x] = sd[0];
}

// Exclusive scan of block sums (single block; nb <= 256).
__global__ void scan_top(int* __restrict__ bsum, int nb) {
  __shared__ int sd[256];
  const int t = threadIdx.x;
  const int orig = (t < nb) ? bsum[t] : 0;
  sd[t] = orig;
  __syncthreads();
  for (int off = 1; off < 256; off <<= 1) {
    const int v = (t >= off) ? sd[t - off] : 0;
    __syncthreads();
    sd[t] += v;
    __syncthreads();
  }
  if (t < nb) bsum[t] = sd[t] - orig;
}

// Per-chunk exclusive scan + block offset -> rowptr; rowptr[n] = total.
__global__ void scan_apply(const int* __restrict__ cnt, int n,
                           const int* __restrict__ bsum,
                           int* __restrict__ rowptr, int total) {
  __shared__ int sd[256];
  const int t = threadIdx.x;
  const int base = blockIdx.x * 1024 + t * 4;
  int v[4];
#pragma unroll
  for (int j = 0; j < 4; ++j) v[j] = (base + j < n) ? cnt[base + j] : 0;
  const int tsum = v[0] + v[1] + v[2] + v[3];
  sd[t] = tsum;
  __syncthreads();
  for (int off = 1; off < 256; off <<= 1) {
    const int u = (t >= off) ? sd[t - off] : 0;
    __syncthreads();
    sd[t] += u;
    __syncthreads();
  }
  int run = sd[t] - tsum + bsum[blockIdx.x];
#pragma unroll
  for (int j = 0; j < 4; ++j) {
    if (base + j < n) rowptr[base + j] = run;
    run += v[j];
  }
  if (blockIdx.x == 0 && t == 0) rowptr[n] = total;
}

__global__ void csr_fill(const int* __restrict__ ei, int E, int total,
                         const int* __restrict__ rowptr,
                         int* __restrict__ cursor, int* __restrict__ csrc) {
  const int e = blockIdx.x * blockDim.x + threadIdx.x;
  if (e >= total) return;
  int s, d;
  if (e < E) { s = ei[e]; d = ei[E + e]; }
  else       { s = e - E; d = e - E; }
  const int pos = rowptr[d] + atomicAdd(&cursor[d], 1);
  csrc[pos] = s;
}

// ---------------------------------------------------------------------------
// Fused segment softmax + aggregation, atomic-free:
//   Y[n,h,:] = (1/(sum_e w_e + 1e-16)) * sum_e w_e * H[src_e,h,:],
//   w_e = exp(lrelu(als[src]+ald[n]) - max_e).
// One thread per (node, head); C-channel accumulators in registers.
// ---------------------------------------------------------------------------
template <int C>
__global__ void gat_agg(const int* __restrict__ rowptr,
                        const int* __restrict__ csrc,
                        const float* __restrict__ als,
                        const float* __restrict__ ald,
                        const float* __restrict__ H,
                        float* __restrict__ Y, int n) {
  const int i = blockIdx.x * blockDim.x + threadIdx.x;
  if (i >= n * NHEADS) return;
  const int node = i >> 2, hd = i & 3;
  const int beg = rowptr[node], end = rowptr[node + 1];
  const float ad = ald[i];  // ald[node*4 + hd]

  float m = -3.402823466e38f;  // every node has a self loop -> deg >= 1
  for (int p = beg; p < end; ++p)
    m = fmaxf(m, lrelu(als[csrc[p] * NHEADS + hd] + ad));

  float acc[C];
#pragma unroll
  for (int c = 0; c < C; ++c) acc[c] = 0.f;
  float ssum = 0.f;

  for (int p = beg; p < end; ++p) {
    const int s = csrc[p];
    const float w = __expf(lrelu(als[s * NHEADS + hd] + ad) - m);
    ssum += w;
    const float4* hs = (const float4*)(H + (size_t)s * NHEADS * C + hd * C);
#pragma unroll
    for (int c4 = 0; c4 < C / 4; ++c4) {
      const float4 hv = hs[c4];
      acc[4 * c4 + 0] += w * hv.x;
      acc[4 * c4 + 1] += w * hv.y;
      acc[4 * c4 + 2] += w * hv.z;
      acc[4 * c4 + 3] += w * hv.w;
    }
  }

  const float inv = 1.f / (ssum + 1e-16f);
  float4* yd = (float4*)(Y + (size_t)node * NHEADS * C + hd * C);
#pragma unroll
  for (int c4 = 0; c4 < C / 4; ++c4) {
    float4 o;
    o.x = acc[4 * c4 + 0] * inv;
    o.y = acc[4 * c4 + 1] * inv;
    o.z = acc[4 * c4 + 2] * inv;
    o.w = acc[4 * c4 + 3] * inv;
    yd[c4] = o;
  }
}

// ---------------------------------------------------------------------------
// Column statistics of (X[:,c] + bias[c]); blockDim.x multiple of D.
// ---------------------------------------------------------------------------
__global__ void col_stats(const float* __restrict__ X, int ld,
                          const float* __restrict__ bias,
                          float* __restrict__ sums, float* __restrict__ sumsq,
                          int n, int D) {
  const int col = threadIdx.x % D;
  const int rpb = blockDim.x / D;
  const int r0  = blockIdx.x * rpb + threadIdx.x / D;
  const int rs  = gridDim.x * rpb;
  const float b = bias ? bias[col] : 0.f;
  float s = 0.f, sq = 0.f;
  for (int r = r0; r < n; r += rs) {
    const float v = X[(size_t)r * ld + col] + b;
    s += v; sq += v * v;
  }
  atomicAdd(&sums[col], s);
  atomicAdd(&sumsq[col], sq);
}

__global__ void bn_apply(const float* __restrict__ X, int ld,
                         const float* __restrict__ bias,
                         const float* __restrict__ sums,
                         const float* __restrict__ sumsq,
                         const float* __restrict__ g,
                         const float* __restrict__ be,
                         float* __restrict__ Y, int n, int D, int relu) {
  const int i = blockIdx.x * blockDim.x + threadIdx.x;
  if (i >= n * D) return;
  const int r = i / D, c = i - r * D;
  const float inv_n = 1.0f / (float)n;
  const float mu = sums[c] * inv_n;
  const float var = sumsq[c] * inv_n - mu * mu;
  float v = X[(size_t)r * ld + c] + (bias ? bias[c] : 0.f);
  v = (v - mu) * rsqrtf(var + 1e-5f) * g[c] + be[c];
  if (relu) v = fmaxf(v, 0.f);
  Y[(size_t)r * D + c] = v;
}

__global__ void final_linear(const float* __restrict__ H,
                             const float* __restrict__ ow,
                             const float* __restrict__ ob,
                             float* __restrict__ out, int n) {
  const int i = blockIdx.x * blockDim.x + threadIdx.x;
  if (i >= n) return;
  float acc = ob[0];
  const float4* hp = (const float4*)(H + (size_t)i * 32);
#pragma unroll
  for (int c4 = 0; c4 < 8; ++c4) {
    const float4 h = hp[c4];
    const float4 w = ((const float4*)ow)[c4];
    acc += h.x * w.x + h.y * w.y + h.z * w.z + h.w * w.w;
  }
  out[i] = acc;
}

// ---------------------------------------------------------------------------
// Host orchestration
// ---------------------------------------------------------------------------
static inline int cdiv(long long a, int b) { return (int)((a + b - 1) / b); }

extern "C" void kernel_launch(void* const* d_in, const int* in_sizes, int n_in,
                              void* d_out, int out_size, void* d_ws, size_t ws_size,
                              hipStream_t stream) {
  const int N = in_sizes[0] / 22;  // 100000
  const int E = in_sizes[1] / 2;   // 1600000
  const int total = E + N;         // with self loops

  const float* x  = (const float*)d_in[0];
  const int*   ei = (const int*)d_in[1];
  const float *Wl[4], *asl[4], *adl[4], *bl[4], *gl[4], *bel[4];
  for (int l = 0; l < 4; ++l) {
    Wl[l]  = (const float*)d_in[2 + 6 * l];
    asl[l] = (const float*)d_in[3 + 6 * l];
    adl[l] = (const float*)d_in[4 + 6 * l];
    bl[l]  = (const float*)d_in[5 + 6 * l];
    gl[l]  = (const float*)d_in[6 + 6 * l];
    bel[l] = (const float*)d_in[7 + 6 * l];
  }
  const float* g0  = (const float*)d_in[26];
  const float* be0 = (const float*)d_in[27];
  const float* ow  = (const float*)d_in[28];
  const float* ob  = (const float*)d_in[29];

  static const int DOUT[4] = {64, 128, 64, 32};

  // Workspace layout
  float* ws    = (float*)d_ws;
  float* xb    = ws;                          // N*4
  float* bufA  = xb + (size_t)N * 4;          // N*128
  float* bufB  = bufA + (size_t)N * 128;      // N*128
  float* als   = bufB + (size_t)N * 128;      // N*4
  float* ald   = als + (size_t)N * 4;         // N*4
  float* sums  = ald + (size_t)N * 4;         // 128
  float* sumsq = sums + 128;                  // 128
  int* cnt     = (int*)(sumsq + 128);         // N  (reused as scatter cursor)
  int* rowptr  = cnt + N;                     // N+1
  int* csrc    = rowptr + (N + 1);            // total
  int* bsum    = csrc + total;                // <=256

  const int T = 256;
  const int NB = cdiv(N, 1024);  // 98 blocks for the scan (<= 256 required)

  // ---- Build CSR by destination (shared by all 4 layers) ----
  hipMemsetAsync(cnt, 0, (size_t)N * sizeof(int), stream);
  deg_count<<<cdiv(total, T), T, 0, stream>>>(ei, E, total, cnt);
  scan_partial<<<NB, 256, 0, stream>>>(cnt, N, bsum);
  scan_top<<<1, 256, 0, stream>>>(bsum, NB);
  scan_apply<<<NB, 256, 0, stream>>>(cnt, N, bsum, rowptr, total);
  hipMemsetAsync(cnt, 0, (size_t)N * sizeof(int), stream);  // -> cursor
  csr_fill<<<cdiv(total, T), T, 0, stream>>>(ei, E, total, rowptr, cnt, csrc);

  // ---- Input BN on x[:, :4] (ld = 22), no bias, no relu ----
  hipMemsetAsync(sums, 0, 2 * 128 * sizeof(float), stream);
  col_stats<<<256, T, 0, stream>>>(x, 22, nullptr, sums, sumsq, N, 4);
  bn_apply<<<cdiv((long long)N * 4, T), T, 0, stream>>>(
      x, 22, nullptr, sums, sumsq, g0, be0, xb, N, 4, /*relu=*/0);

  // ---- 4 GAT layers ----
  float* bufs[2] = {bufA, bufB};
  const float* in = xb;
  int pp = 0;
  for (int l = 0; l < 4; ++l) {
    const int dout = DOUT[l], C = dout / NHEADS;
    float* h = bufs[pp];
    float* Y = (l == 0) ? bufs[1] : (float*)in;  // gemm reads `in` first

    // h = in @ W  (WMMA, W staged in LDS)
    const int gB = cdiv(N / 16, 4);  // 4 waves per block, 1 M-tile per wave
    switch (l) {
      case 0: gemm_wmma_f32<4, 64><<<gB, 128, 0, stream>>>(in, Wl[l], h, N); break;
      case 1: gemm_wmma_f32<64, 128><<<gB, 128, 0, stream>>>(in, Wl[l], h, N); break;
      case 2: gemm_wmma_f32<128, 64><<<gB, 128, 0, stream>>>(in, Wl[l], h, N); break;
      case 3: gemm_wmma_f32<64, 32><<<gB, 128, 0, stream>>>(in, Wl[l], h, N); break;
    }

    // attention logits
    gat_scores<<<cdiv((long long)N * NHEADS, T), T, 0, stream>>>(
        h, asl[l], adl[l], als, ald, N, C);

    // fused segment softmax + aggregation (no atomics, overwrites Y fully)
    const int gA = cdiv((long long)N * NHEADS, T);
    switch (C) {
      case 8:  gat_agg<8><<<gA, T, 0, stream>>>(rowptr, csrc, als, ald, h, Y, N); break;
      case 16: gat_agg<16><<<gA, T, 0, stream>>>(rowptr, csrc, als, ald, h, Y, N); break;
      case 32: gat_agg<32><<<gA, T, 0, stream>>>(rowptr, csrc, als, ald, h, Y, N); break;
    }

    // BN(Y + bias) with gamma/beta, relu; write result over h
    hipMemsetAsync(sums, 0, 2 * 128 * sizeof(float), stream);
    col_stats<<<256, T, 0, stream>>>(Y, dout, bl[l], sums, sumsq, N, dout);
    bn_apply<<<cdiv((long long)N * dout, T), T, 0, stream>>>(
        Y, dout, bl[l], sums, sumsq, gl[l], bel[l], h, N, dout, /*relu=*/1);

    in = h;
    pp ^= 1;
  }

  // ---- Final linear [N,32] @ [32,1] + ob ----
  final_linear<<<cdiv(N, T), T, 0, stream>>>(in, ow, ob, (float*)d_out, N);
}